// RadiusInteractionGraph_30812095381892
// MI455X (gfx1250) — compile-verified
//
#include <hip/hip_runtime.h>
#include <hip/hip_bf16.h>

typedef __attribute__((ext_vector_type(2))) float v2f;
typedef __attribute__((ext_vector_type(4))) float f4;
typedef __attribute__((ext_vector_type(8))) float v8f;

#define B_MOLS     4096
#define N_ATOMS    64
#define KNBR       32
#define CUT2       100.0f
#define ROW_STRIDE 68   // 16B-aligned rows; at worst 2-way LDS bank conflict

// One block = one molecule. 64 threads = 2 wave32.
// Phase 1: stage padded positions (x,y,z,0) + squared norms in LDS.
// Phase 2: Gram matrix P*P^T via v_wmma_f32_16x16x4_f32 (K=4 = xyz+pad),
//          epilogue sq = |xi|^2 + |xj|^2 - 2*Gij -> LDS.
// Phase 3: register bitonic TOP-32: sort 32-blocks (asc/desc), pairwise min,
//          bitonic-merge the surviving 32 -> ascending.
// Phase 4: emit src/dst/weight as 8x b128 non-temporal stores per array.
__global__ __launch_bounds__(64)
void radius_graph_kernel(const float* __restrict__ pos,
                         float* __restrict__ out,
                         const int E) {
    __shared__ float sP[N_ATOMS * 4];           // padded positions
    __shared__ float sN[N_ATOMS];               // squared norms
    __shared__ float sD[N_ATOMS * ROW_STRIDE];  // squared distances

    const int m    = blockIdx.x;
    const int t    = threadIdx.x;   // 0..63, also "my row"
    const int lane = t & 31;
    const int wave = t >> 5;

    // ---- Phase 1: load positions, pad K=4, compute norms ----
    {
        const float* p = pos + (size_t)(m * N_ATOMS) * 3;
        const float x = p[t * 3 + 0];
        const float y = p[t * 3 + 1];
        const float z = p[t * 3 + 2];
        sP[t * 4 + 0] = x;
        sP[t * 4 + 1] = y;
        sP[t * 4 + 2] = z;
        sP[t * 4 + 3] = 0.0f;
        sN[t] = x * x + y * y + z * z;
    }
    __syncthreads();

    // ---- Phase 2: Gram via WMMA f32 16x16x4 ----
    // A (16x4): lanes 0-15 -> M=lane, K={0,1}; lanes 16-31 -> M=lane-16, K={2,3}.
    // B (4x16) mirrors A. D (16x16): lanes 0-15 -> M=r, N=lane; 16-31 -> M=r+8, N=lane-16.
    {
        const int sub   = lane & 15;
        const int khalf = (lane >> 4) * 2;
        #pragma unroll
        for (int ti = 0; ti < 2; ++ti) {
            const int I = wave * 2 + ti;   // row tile 0..3
            v2f a;
            a.x = sP[(I * 16 + sub) * 4 + khalf + 0];
            a.y = sP[(I * 16 + sub) * 4 + khalf + 1];
            #pragma unroll
            for (int J = 0; J < 4; ++J) {  // col tile 0..3
                v2f b;
                b.x = sP[(J * 16 + sub) * 4 + khalf + 0];
                b.y = sP[(J * 16 + sub) * 4 + khalf + 1];
                v8f c = {};
                c = __builtin_amdgcn_wmma_f32_16x16x4_f32(
                        /*neg_a=*/false, a, /*neg_b=*/false, b,
                        /*c_mod=*/(short)0, c,
                        /*reuse_a=*/false, /*reuse_b=*/false);
                const int Mbase = I * 16 + 8 * (lane >> 4);
                const int Nn    = J * 16 + sub;
                const float nn  = sN[Nn];
                #pragma unroll
                for (int r = 0; r < 8; ++r) {
                    const int M = Mbase + r;
                    sD[M * ROW_STRIDE + Nn] = sN[M] + nn - 2.0f * c[r];
                }
            }
        }
    }
    __syncthreads();

    // ---- Phase 3: pack sortable keys (b128 row loads), bitonic top-32 ----
    unsigned key[N_ATOMS];
    {
        const f4* row = (const f4*)&sD[t * ROW_STRIDE];
        #pragma unroll
        for (int q = 0; q < 16; ++q) {
            const f4 v = row[q];
            #pragma unroll
            for (int c = 0; c < 4; ++c) {
                const int j = q * 4 + c;
                const float sq = v[c];
                const bool valid = (j != t) && (sq > 0.0f) && (sq < CUT2);
                // sq >= 0 -> IEEE bits order-monotone; low 6 bits carry index
                const unsigned packed = (__float_as_uint(sq) & 0xFFFFFFC0u) | (unsigned)j;
                key[j] = valid ? packed : (0x7F800000u | (unsigned)j);  // +inf | j
            }
        }
    }
    // Stages k=2..32: block [0..31] ascending, block [32..63] descending.
    #pragma unroll
    for (int k = 2; k <= 32; k <<= 1) {
        #pragma unroll
        for (int j = k >> 1; j > 0; j >>= 1) {
            #pragma unroll
            for (int i = 0; i < N_ATOMS; ++i) {
                const int ixj = i ^ j;
                if (ixj > i) {
                    const bool up = ((i & k) == 0);
                    const unsigned a0 = key[i];
                    const unsigned b0 = key[ixj];
                    const bool sw = up ? (a0 > b0) : (a0 < b0);
                    key[i]   = sw ? b0 : a0;
                    key[ixj] = sw ? a0 : b0;
                }
            }
        }
    }
    // k=64, j=32 round, min side only: lower half = 32 smallest (bitonic).
    #pragma unroll
    for (int i = 0; i < KNBR; ++i) {
        const unsigned b0 = key[i + 32];
        key[i] = (key[i] < b0) ? key[i] : b0;
    }
    // Bitonic merge of 32, ascending.
    #pragma unroll
    for (int j = 16; j > 0; j >>= 1) {
        #pragma unroll
        for (int i = 0; i < KNBR; ++i) {
            const int ixj = i ^ j;
            if (ixj > i) {
                const unsigned a0 = key[i];
                const unsigned b0 = key[ixj];
                const bool sw = a0 > b0;
                key[i]   = sw ? b0 : a0;
                key[ixj] = sw ? a0 : b0;
            }
        }
    }

    // ---- Phase 4: emit edges as b128 non-temporal stores ----
    float* out_src = out;
    float* out_dst = out + (size_t)E;
    float* out_w   = out + (size_t)2 * E;
    const int   base   = (m * N_ATOMS + t) * KNBR;   // 128B-aligned
    const int   gbase  = m * N_ATOMS;
    const float dstIdx = (float)(gbase + t);
    #pragma unroll
    for (int q = 0; q < KNBR / 4; ++q) {
        f4 vs, vd, vw;
        #pragma unroll
        for (int c = 0; c < 4; ++c) {
            const int s = q * 4 + c;
            const int j = (int)(key[s] & 63u);
            vs[c] = (float)(gbase + j);
            vd[c] = dstIdx;
            vw[c] = sqrtf(sD[t * ROW_STRIDE + j]);
        }
        __builtin_nontemporal_store(vs, (f4*)(out_src + base) + q);
        __builtin_nontemporal_store(vd, (f4*)(out_dst + base) + q);
        __builtin_nontemporal_store(vw, (f4*)(out_w   + base) + q);
    }
}

extern "C" void kernel_launch(void* const* d_in, const int* in_sizes, int n_in,
                              void* d_out, int out_size, void* d_ws, size_t ws_size,
                              hipStream_t stream) {
    const float* pos = (const float*)d_in[0];
    // d_in[1] = batch ids (int32): structure is uniform (64 atoms/molecule), unused.
    (void)in_sizes; (void)n_in; (void)d_ws; (void)ws_size; (void)out_size;

    float* out = (float*)d_out;
    const int E = B_MOLS * N_ATOMS * KNBR;   // 8,388,608 edges

    radius_graph_kernel<<<dim3(B_MOLS), dim3(64), 0, stream>>>(pos, out, E);
}